// GaussianRasterizer_81466939670818
// MI455X (gfx1250) — compile-verified
//
#include <hip/hip_runtime.h>
#include <stdint.h>

#define NG    2048
#define HH    96
#define WW    96
#define NPIX  (HH*WW)
#define FXc   96.0f   // W / (2*TANX)
#define FYc   96.0f

typedef __attribute__((ext_vector_type(2))) float v2f;
typedef __attribute__((ext_vector_type(8))) float v8f;

// workspace layout (float indices)
#define O_CA 0
#define O_CB 2048
#define O_CC 4096
#define O_PX 6144
#define O_PY 8192
#define O_OP 10240
#define O_R  12288
#define O_G  14336
#define O_B  16384
#define O_D  18432
#define O_S  20480   // 9 sorted arrays follow, each 2048 floats

// ---------------------------------------------------------------- preprocess
__global__ void gs_preprocess(const float* __restrict__ means,
                              const float* __restrict__ cols,
                              const float* __restrict__ opac,
                              const float* __restrict__ scales,
                              const float* __restrict__ rots,
                              const float* __restrict__ viewm,
                              const float* __restrict__ projm,
                              float* __restrict__ ws,
                              int* __restrict__ radii_out)
{
    int i = blockIdx.x * blockDim.x + threadIdx.x;
    if (i >= NG) return;

    float mx = means[3*i+0], my = means[3*i+1], mz = means[3*i+2];

    // p_view = view @ [m,1]
    float pv0 = viewm[0]*mx + viewm[1]*my + viewm[2]*mz  + viewm[3];
    float pv1 = viewm[4]*mx + viewm[5]*my + viewm[6]*mz  + viewm[7];
    float pv2 = viewm[8]*mx + viewm[9]*my + viewm[10]*mz + viewm[11];
    float depth = pv2;

    // p_hom = proj @ [m,1] ; p_proj = p_hom/(w+1e-7)
    float h0 = projm[0]*mx  + projm[1]*my  + projm[2]*mz  + projm[3];
    float h1 = projm[4]*mx  + projm[5]*my  + projm[6]*mz  + projm[7];
    float hw = projm[12]*mx + projm[13]*my + projm[14]*mz + projm[15];
    float invw = 1.0f / (hw + 1e-7f);
    float ppx = h0 * invw, ppy = h1 * invw;

    // quaternion -> rotation
    float q0 = rots[4*i+0], q1 = rots[4*i+1], q2 = rots[4*i+2], q3 = rots[4*i+3];
    float qn = rsqrtf(q0*q0 + q1*q1 + q2*q2 + q3*q3);
    float r = q0*qn, x = q1*qn, y = q2*qn, z = q3*qn;
    float R00 = 1.0f-2.0f*(y*y+z*z), R01 = 2.0f*(x*y-r*z), R02 = 2.0f*(x*z+r*y);
    float R10 = 2.0f*(x*y+r*z), R11 = 1.0f-2.0f*(x*x+z*z), R12 = 2.0f*(y*z-r*x);
    float R20 = 2.0f*(x*z-r*y), R21 = 2.0f*(y*z+r*x), R22 = 1.0f-2.0f*(x*x+y*y);

    float s0 = scales[3*i+0], s1 = scales[3*i+1], s2 = scales[3*i+2];
    float M00=R00*s0, M01=R01*s1, M02=R02*s2;
    float M10=R10*s0, M11=R11*s1, M12=R12*s2;
    float M20=R20*s0, M21=R21*s1, M22=R22*s2;

    // Sigma = M M^T (symmetric)
    float S00 = M00*M00 + M01*M01 + M02*M02;
    float S01 = M00*M10 + M01*M11 + M02*M12;
    float S02 = M00*M20 + M01*M21 + M02*M22;
    float S11 = M10*M10 + M11*M11 + M12*M12;
    float S12 = M10*M20 + M11*M21 + M12*M22;
    float S22 = M20*M20 + M21*M21 + M22*M22;

    // J (2x3)
    float tz = depth;
    float invz = 1.0f / tz;
    float txl = fminf(fmaxf(pv0*invz, -0.65f), 0.65f) * tz;  // 1.3*TANX = 0.65
    float tyl = fminf(fmaxf(pv1*invz, -0.65f), 0.65f) * tz;
    float J00 = FXc*invz, J02 = -FXc*txl*invz*invz;
    float J11 = FYc*invz, J12 = -FYc*tyl*invz*invz;

    // Sv = Wr Sigma Wr^T, Wr = view[:3,:3]
    float W00=viewm[0], W01=viewm[1], W02=viewm[2];
    float W10=viewm[4], W11=viewm[5], W12=viewm[6];
    float W20=viewm[8], W21=viewm[9], W22=viewm[10];
    // t = Wr * Sigma
    float t00 = W00*S00 + W01*S01 + W02*S02;
    float t01 = W00*S01 + W01*S11 + W02*S12;
    float t02 = W00*S02 + W01*S12 + W02*S22;
    float t10 = W10*S00 + W11*S01 + W12*S02;
    float t11 = W10*S01 + W11*S11 + W12*S12;
    float t12 = W10*S02 + W11*S12 + W12*S22;
    float t20 = W20*S00 + W21*S01 + W22*S02;
    float t21 = W20*S01 + W21*S11 + W22*S12;
    float t22 = W20*S02 + W21*S12 + W22*S22;
    // Sv = t * Wr^T (symmetric)
    float Sv00 = t00*W00 + t01*W01 + t02*W02;
    float Sv01 = t00*W10 + t01*W11 + t02*W12;
    float Sv02 = t00*W20 + t01*W21 + t02*W22;
    float Sv11 = t10*W10 + t11*W11 + t12*W12;
    float Sv12 = t10*W20 + t11*W21 + t12*W22;
    float Sv22 = t20*W20 + t21*W21 + t22*W22;

    // cov2d = J Sv J^T  (J10 = J01 = 0)
    float JS00 = J00*Sv00 + J02*Sv02;
    float JS01 = J00*Sv01 + J02*Sv12;
    float JS02 = J00*Sv02 + J02*Sv22;
    float JS11 = J11*Sv11 + J12*Sv12;
    float JS12 = J11*Sv12 + J12*Sv22;
    float a = JS00*J00 + JS02*J02 + 0.3f;
    float b = JS01*J11 + JS02*J12;
    float c = JS11*J11 + JS12*J12 + 0.3f;

    float det = a*c - b*b;
    bool valid = (depth > 0.2f) && (det > 0.0f);
    float inv_det = (det != 0.0f) ? (1.0f/det) : 0.0f;
    float conA =  c * inv_det;
    float conB = -b * inv_det;
    float conC =  a * inv_det;

    float mid = 0.5f*(a + c);
    float lam = mid + sqrtf(fmaxf(0.1f, mid*mid - det));
    int radii = valid ? (int)ceilf(3.0f * sqrtf(lam)) : 0;

    float px = ((ppx + 1.0f) * (float)WW - 1.0f) * 0.5f;
    float py = ((ppy + 1.0f) * (float)HH - 1.0f) * 0.5f;

    ws[O_CA + i] = conA;
    ws[O_CB + i] = conB;
    ws[O_CC + i] = conC;
    ws[O_PX + i] = px;
    ws[O_PY + i] = py;
    ws[O_OP + i] = valid ? opac[i] : 0.0f;   // invalid -> alpha 0
    ws[O_R  + i] = cols[3*i+0];
    ws[O_G  + i] = cols[3*i+1];
    ws[O_B  + i] = cols[3*i+2];
    ws[O_D  + i] = depth;
    radii_out[i] = radii;
}

// ------------------------------------------------- bitonic sort by depth + gather
__global__ __launch_bounds__(1024)
void gs_sort_gather(float* __restrict__ ws)
{
    __shared__ unsigned long long sk[NG];   // 16 KB
    int t = threadIdx.x;

    for (int i = t; i < NG; i += 1024) {
        float d = ws[O_D + i];
        unsigned int fb = __float_as_uint(d);
        unsigned int key = (fb & 0x80000000u) ? ~fb : (fb | 0x80000000u);
        sk[i] = ((unsigned long long)key << 32) | (unsigned int)i;
    }
    __syncthreads();

    for (int k = 2; k <= NG; k <<= 1) {
        for (int j = k >> 1; j > 0; j >>= 1) {
            for (int i = t; i < NG; i += 1024) {
                int ixj = i ^ j;
                if (ixj > i) {
                    bool up = ((i & k) == 0);
                    unsigned long long va = sk[i], vb = sk[ixj];
                    if ((va > vb) == up) { sk[i] = vb; sk[ixj] = va; }
                }
            }
            __syncthreads();
        }
    }

    // gather all params into depth-sorted order (struct-of-arrays)
    for (int i = t; i < NG; i += 1024) {
        int si = (int)(sk[i] & 0xFFFFFFFFull);
        ws[O_S + 0*NG + i] = ws[O_CA + si];
        ws[O_S + 1*NG + i] = ws[O_CB + si];
        ws[O_S + 2*NG + i] = ws[O_CC + si];
        ws[O_S + 3*NG + i] = ws[O_PX + si];
        ws[O_S + 4*NG + i] = ws[O_PY + si];
        ws[O_S + 5*NG + i] = ws[O_OP + si];
        ws[O_S + 6*NG + i] = ws[O_R  + si];
        ws[O_S + 7*NG + i] = ws[O_G  + si];
        ws[O_S + 8*NG + i] = ws[O_B  + si];
    }
}

// ---------------------------------------------------------------- rasterize
// One wave32 owns 16 consecutive pixels. C (16x16 f32) = pixels x channels.
// Per chunk of 4 sorted Gaussians: A = wgt (16x4), B = colors (4x16),
// C += A*B via V_WMMA_F32_16X16X4_F32.  Two chunks (K=8) per loop trip.
// A layout: lanes 0-15 -> M=lane, {K0,K1}; lanes 16-31 -> M=lane-16, {K2,K3}.
// Every lane tracks transmittance T for pixel m = lane&15 (duplicated in the
// two half-waves, which keeps the scan cross-lane-free).
__global__ __launch_bounds__(256)
void gs_raster(const float* __restrict__ ws,
               const float* __restrict__ bg,
               float* __restrict__ out)
{
    int gtid = blockIdx.x * blockDim.x + threadIdx.x;
    int wave = gtid >> 5;
    int lane = gtid & 31;
    bool hi  = lane >= 16;
    int base = wave * 16;          // first pixel of this wave's tile
    int m    = lane & 15;          // local pixel index this lane scans
    int p    = base + m;
    float fgx = (float)(p % WW);
    float fgy = (float)(p / WW);

    const float* cA  = ws + O_S + 0*NG;
    const float* cB  = ws + O_S + 1*NG;
    const float* cC  = ws + O_S + 2*NG;
    const float* cPX = ws + O_S + 3*NG;
    const float* cPY = ws + O_S + 4*NG;
    const float* cOP = ws + O_S + 5*NG;
    const float* cR  = ws + O_S + 6*NG;
    const float* cG  = ws + O_S + 7*NG;
    const float* cBl = ws + O_S + 8*NG;

    // loop-invariant, branchless channel selectors for the B matrix
    int ch = lane & 15;
    float selR = (ch == 0) ? 1.0f : 0.0f;
    float selG = (ch == 1) ? 1.0f : 0.0f;
    float selB = (ch == 2) ? 1.0f : 0.0f;

    float T = 1.0f;
    v8f acc = {};

    for (int c8 = 0; c8 < NG/8; ++c8) {
#pragma unroll
        for (int hlf = 0; hlf < 2; ++hlf) {
            int g0 = c8 * 8 + hlf * 4;
            float w[4], col[4];
#pragma unroll
            for (int k = 0; k < 4; ++k) {
                int g = g0 + k;                 // uniform -> scalar loads
                float A  = cA[g],  B  = cB[g],  C  = cC[g];
                float PX = cPX[g], PY = cPY[g], OP = cOP[g];
                float rr = cR[g],  gg = cG[g],  bb = cBl[g];
                float dx = PX - fgx, dy = PY - fgy;
                float pw = -0.5f*(A*dx*dx + C*dy*dy) - B*dx*dy;
                float al = fminf(0.99f, OP * __expf(fminf(pw, 0.0f)));
                al = (pw <= 0.0f && al >= 0.003921568627f) ? al : 0.0f;
                w[k] = al * T;
                T *= (1.0f - al);
                col[k] = fmaf(selR, rr, fmaf(selG, gg, selB * bb));
            }
            v2f Am = { hi ? w[2]   : w[0],   hi ? w[3]   : w[1]   };
            v2f Bm = { hi ? col[2] : col[0], hi ? col[3] : col[1] };
            acc = __builtin_amdgcn_wmma_f32_16x16x4_f32(
                      false, Am, false, Bm, (short)0, acc, false, false);
        }
    }

    // epilogue: out[ch, base+M] = acc[M][ch] + bg[ch]*T_final[M]
    float bgc = fmaf(selR, bg[0], fmaf(selG, bg[1], selB * bg[2]));
#pragma unroll
    for (int r2 = 0; r2 < 8; ++r2) {
        int M = r2 + (hi ? 8 : 0);
        float Tm = __shfl(T, M, 32);        // T for pixel M lives in lane M
        if (ch < 3)
            out[ch * NPIX + base + M] = acc[r2] + bgc * Tm;
    }
}

// ---------------------------------------------------------------- launcher
extern "C" void kernel_launch(void* const* d_in, const int* in_sizes, int n_in,
                              void* d_out, int out_size, void* d_ws, size_t ws_size,
                              hipStream_t stream)
{
    const float* means  = (const float*)d_in[0];
    const float* cols   = (const float*)d_in[1];
    const float* opac   = (const float*)d_in[2];
    const float* scales = (const float*)d_in[3];
    const float* rots   = (const float*)d_in[4];
    const float* viewm  = (const float*)d_in[5];
    const float* projm  = (const float*)d_in[6];
    const float* bg     = (const float*)d_in[8];   // d_in[7] = campos, unused

    float* ws  = (float*)d_ws;
    float* out = (float*)d_out;
    int* radii = (int*)d_out + NPIX * 3;           // radii int32 bits after color

    hipLaunchKernelGGL(gs_preprocess, dim3(NG/256), dim3(256), 0, stream,
                       means, cols, opac, scales, rots, viewm, projm, ws, radii);
    hipLaunchKernelGGL(gs_sort_gather, dim3(1), dim3(1024), 0, stream, ws);
    hipLaunchKernelGGL(gs_raster, dim3((NPIX/16)/8), dim3(256), 0, stream,
                       ws, bg, out);
}